// LGCN_Encoder_62105227100257
// MI455X (gfx1250) — compile-verified
//
#include <hip/hip_runtime.h>
#include <stdint.h>

#define USER_NUM 100000
#define ITEM_NUM 50000
#define N_NODES  150000
#define EMB      128
#define N_EDGES  600000
#define RPB      8          // rows per block in spmm (8 waves * 32 lanes = 256 thr)
#define EDGE_CAP 512        // LDS staging capacity (edges); avg block load = 32

typedef unsigned int u32;
typedef unsigned long long u64;

// ---------------------------------------------------------------------------
// Tensor Data Mover staging (CDNA5 / gfx1250). This toolchain exposes the
// 6-arg builtin: (u32x4 g0, i32x8 g1, i32x4 g2, i32x4 g3, i32x8 extra, i32 cpol)
// ---------------------------------------------------------------------------
#if defined(__gfx1250__) && __has_builtin(__builtin_amdgcn_tensor_load_to_lds) && \
    __has_builtin(__builtin_amdgcn_s_wait_tensorcnt)
#define HAVE_TDM 1
typedef u32 u32x4 __attribute__((ext_vector_type(4)));
typedef int i32x4 __attribute__((ext_vector_type(4)));
typedef int i32x8 __attribute__((ext_vector_type(8)));

// 1-D contiguous tile load: nelem 4-byte elements from gsrc to LDS at byte offset lds_off.
__device__ __forceinline__ void tdm_load_1d(const void* gsrc, u32 lds_off, u32 nelem) {
  u64 ga = (u64)(uintptr_t)gsrc;
  u32x4 g0;
  g0[0] = 1u;                                              // count=1 (valid user D#)
  g0[1] = lds_off;                                         // lds_addr (bytes)
  g0[2] = (u32)ga;                                         // global_addr[31:0]
  g0[3] = (u32)((ga >> 32) & 0x1FFFFFFull) | (2u << 30);   // addr[56:32] | type=2
  i32x8 g1;
  g1[0] = (int)(2u << 16);                                 // data_size = 4B
  g1[1] = (int)((nelem & 0xFFFFu) << 16);                  // tensor_dim0[15:0]
  g1[2] = (int)(((nelem >> 16) & 0xFFFFu) | (1u << 16));   // dim0 hi | tensor_dim1=1
  g1[3] = (int)((nelem & 0xFFFFu) << 16);                  // tile_dim0 = nelem
  g1[4] = 1;                                               // tile_dim1 = 1
  g1[5] = (int)nelem;                                      // tensor_dim0_stride
  g1[6] = 0;
  g1[7] = 0;
  i32x4 z4 = {0, 0, 0, 0};
  i32x8 z8 = {0, 0, 0, 0, 0, 0, 0, 0};
  __builtin_amdgcn_tensor_load_to_lds(g0, g1, z4, z4, z8, 0);
}
#endif

// ---------------------------------------------------------------------------
// CSR construction kernels (one-time per call; removes ALL float atomics
// from the 3 propagation layers)
// ---------------------------------------------------------------------------
__global__ void zero_i32(int* __restrict__ p, int n) {
  int i = blockIdx.x * blockDim.x + threadIdx.x;
  if (i < n) p[i] = 0;
}

__global__ void hist_k(const int* __restrict__ erow, int* __restrict__ hist) {
  int e = blockIdx.x * blockDim.x + threadIdx.x;
  if (e < N_EDGES) atomicAdd(&hist[erow[e]], 1);
}

// Single-block chunked Hillis-Steele exclusive scan over N_NODES counts.
__global__ void __launch_bounds__(1024) scan_k(const int* __restrict__ hist,
                                               int* __restrict__ row_ptr,
                                               int* __restrict__ cursor) {
  __shared__ int sh[1024];
  __shared__ int s_carry;
  int t = threadIdx.x;
  if (t == 0) s_carry = 0;
  __syncthreads();
  for (int base = 0; base < N_NODES; base += 1024) {
    int i = base + t;
    int x = (i < N_NODES) ? hist[i] : 0;
    sh[t] = x;
    __syncthreads();
    for (int ofs = 1; ofs < 1024; ofs <<= 1) {
      int y = (t >= ofs) ? sh[t - ofs] : 0;
      __syncthreads();
      sh[t] += y;
      __syncthreads();
    }
    int incl = sh[t];
    int carry = s_carry;
    if (i < N_NODES) {
      int ex = carry + incl - x;   // exclusive prefix
      row_ptr[i] = ex;
      cursor[i]  = ex;
    }
    __syncthreads();
    if (t == 1023) s_carry = carry + incl;
    __syncthreads();
  }
  if (t == 0) row_ptr[N_NODES] = s_carry;  // == N_EDGES
}

__global__ void scatter_k(const float* __restrict__ evals,
                          const int* __restrict__ erow,
                          const int* __restrict__ ecol,
                          int* __restrict__ cursor,
                          float* __restrict__ sval,
                          int* __restrict__ scol) {
  int e = blockIdx.x * blockDim.x + threadIdx.x;
  if (e < N_EDGES) {
    int r = erow[e];
    int p = atomicAdd(&cursor[r], 1);
    sval[p] = evals[e];
    scol[p] = ecol[e];
  }
}

// ---------------------------------------------------------------------------
// Fused SpMM layer: wave-per-row CSR gather (no atomics), TDM-staged edge
// data in LDS, with the layer accumulator (and final 1/3 scale) folded in.
// mode 0: acc  = ego_out = spmm(x)
// mode 1: acc += (ego_out = spmm(x))
// mode 2: acc  = (acc + spmm(x)) / 3      (ego_out not written)
// ---------------------------------------------------------------------------
__global__ void __launch_bounds__(RPB * 32) spmm_fused(
    const int* __restrict__ row_ptr, const float* __restrict__ evs,
    const int* __restrict__ ecs, const float* __restrict__ xa,
    const float* __restrict__ xb, int split,
    float* __restrict__ ego_out, float* __restrict__ acc, int mode) {
  __shared__ float s_val[EDGE_CAP];
  __shared__ int   s_col[EDGE_CAP];

  int r0 = blockIdx.x * RPB;
  int rend = r0 + RPB;
  if (rend > N_NODES) rend = N_NODES;
  int e0 = row_ptr[r0];
  int e1 = row_ptr[rend];
  int cnt = e1 - e0;
  bool staged = (cnt > 0) && (cnt <= EDGE_CAP);
  if (staged) {
#if defined(HAVE_TDM)
    int w = threadIdx.x >> 5;
    if (w == 0)      tdm_load_1d(evs + e0, (u32)(uintptr_t)(void*)s_val, (u32)cnt);
    else if (w == 1) tdm_load_1d(ecs + e0, (u32)(uintptr_t)(void*)s_col, (u32)cnt);
    __builtin_amdgcn_s_wait_tensorcnt(0);
#else
    for (int i = threadIdx.x; i < cnt; i += blockDim.x) {
      s_val[i] = evs[e0 + i];
      s_col[i] = ecs[e0 + i];
    }
#endif
  }
  __syncthreads();

  int wave = threadIdx.x >> 5;
  int lane = threadIdx.x & 31;
  int row = r0 + wave;
  if (row >= N_NODES) return;

  int a = row_ptr[row];
  int b = row_ptr[row + 1];
  float4 s = make_float4(0.f, 0.f, 0.f, 0.f);
  for (int e = a; e < b; ++e) {
    float v;
    int c;
    if (staged) { v = s_val[e - e0]; c = s_col[e - e0]; }
    else        { v = evs[e];        c = ecs[e]; }
    const float* base = (c < split) ? (xa + (size_t)c * EMB)
                                    : (xb + (size_t)(c - split) * EMB);
    const float4 f = ((const float4*)base)[lane];   // coalesced 512B row gather
    s.x = fmaf(v, f.x, s.x);
    s.y = fmaf(v, f.y, s.y);
    s.z = fmaf(v, f.z, s.z);
    s.w = fmaf(v, f.w, s.w);
  }

  size_t o = (size_t)row * EMB + (size_t)lane * 4;
  if (mode == 0) {
    *(float4*)(ego_out + o) = s;
    *(float4*)(acc + o) = s;
  } else if (mode == 1) {
    *(float4*)(ego_out + o) = s;
    float4 t = *(float4*)(acc + o);
    t.x += s.x; t.y += s.y; t.z += s.z; t.w += s.w;
    *(float4*)(acc + o) = t;
  } else {
    const float k = 1.0f / 3.0f;
    float4 t = *(float4*)(acc + o);
    t.x = (t.x + s.x) * k;
    t.y = (t.y + s.y) * k;
    t.z = (t.z + s.z) * k;
    t.w = (t.w + s.w) * k;
    *(float4*)(acc + o) = t;
  }
}

// ---------------------------------------------------------------------------
extern "C" void kernel_launch(void* const* d_in, const int* in_sizes, int n_in,
                              void* d_out, int out_size, void* d_ws, size_t ws_size,
                              hipStream_t stream) {
  const float* user_emb = (const float*)d_in[0];
  const float* item_emb = (const float*)d_in[1];
  const float* evals    = (const float*)d_in[2];
  const int*   erow     = (const int*)d_in[3];
  const int*   ecol     = (const int*)d_in[4];
  float* accp = (float*)d_out;   // acc lives in d_out: node order == output order

  char* ws = (char*)d_ws;
  size_t off = 0;
  auto take = [&](size_t bytes) -> void* {
    void* p = ws + off;
    off = (off + bytes + 255) & ~(size_t)255;
    return p;
  };
  float* egoA    = (float*)take((size_t)N_NODES * EMB * sizeof(float));
  float* egoB    = (float*)take((size_t)N_NODES * EMB * sizeof(float));
  int*   hist    = (int*)take((size_t)N_NODES * sizeof(int));
  int*   row_ptr = (int*)take((size_t)(N_NODES + 1) * sizeof(int));
  int*   cursor  = (int*)take((size_t)N_NODES * sizeof(int));
  float* sval    = (float*)take((size_t)N_EDGES * sizeof(float));
  int*   scol    = (int*)take((size_t)N_EDGES * sizeof(int));
  (void)in_sizes; (void)n_in; (void)out_size; (void)ws_size;

  const int eb = (N_EDGES + 255) / 256;
  const int nb = (N_NODES + 255) / 256;
  const int sb = (N_NODES + RPB - 1) / RPB;

  // CSR build (rebuilt each call: deterministic work, no cross-call state)
  zero_i32<<<nb, 256, 0, stream>>>(hist, N_NODES);
  hist_k<<<eb, 256, 0, stream>>>(erow, hist);
  scan_k<<<1, 1024, 0, stream>>>(hist, row_ptr, cursor);
  scatter_k<<<eb, 256, 0, stream>>>(evals, erow, ecol, cursor, sval, scol);

  // Layer 0: gather directly from (user_emb | item_emb); acc = ego1
  spmm_fused<<<sb, 256, 0, stream>>>(row_ptr, sval, scol, user_emb, item_emb,
                                     USER_NUM, egoA, accp, 0);
  // Layer 1: acc += ego2
  spmm_fused<<<sb, 256, 0, stream>>>(row_ptr, sval, scol, egoA, egoA,
                                     N_NODES, egoB, accp, 1);
  // Layer 2: acc = (acc + ego3) / 3
  spmm_fused<<<sb, 256, 0, stream>>>(row_ptr, sval, scol, egoB, egoB,
                                     N_NODES, egoA, accp, 2);
}